// GBST_84894323572774
// MI455X (gfx1250) — compile-verified
//
#include <hip/hip_runtime.h>
#include <hip/hip_bf16.h>
#include <math.h>

typedef float v2f __attribute__((ext_vector_type(2)));
typedef float v8f __attribute__((ext_vector_type(8)));

#define BB 8
#define NN 3072
#define DD 512
#define VOCAB 256
#define MBS 4
#define DSF 4
#define NT (NN / 16)        // 192 tiles of 16 positions
#define GROUPS (NN / DSF)   // 768
#define NCHUNK (NN / 12)    // 256 twelve-position chunks per batch

// workspace layout (in floats)
#define EW_OFF 0            // [256]
#define PW_OFF 256          // [4]
#define S_OFF  512          // [B][N][4]  pre-attention softmax scores
#define S2_OFF (512 + BB * NN * 4)   // [B][N][4] post-attention scores

// ---------------- Kernel A1: ew[v] = dot(token_emb[v], w); pw[j] = dot(pos_emb[j], w)
__global__ void __launch_bounds__(128) k_rowdots(const float* __restrict__ token_emb,
                                                 const float* __restrict__ pos_emb,
                                                 const float* __restrict__ score_w,
                                                 float* __restrict__ ws) {
  __shared__ float red[128];
  int v = blockIdx.x;
  const float* row = (v < VOCAB) ? (token_emb + (size_t)v * DD)
                                 : (pos_emb + (size_t)(v - VOCAB) * DD);
  float acc = 0.f;
  for (int i = threadIdx.x; i < DD; i += 128) acc += row[i] * score_w[i];
  red[threadIdx.x] = acc;
  __syncthreads();
  for (int off = 64; off > 0; off >>= 1) {
    if ((int)threadIdx.x < off) red[threadIdx.x] += red[threadIdx.x + off];
    __syncthreads();
  }
  if (threadIdx.x == 0) {
    if (v < VOCAB) ws[EW_OFF + v] = red[0];
    else           ws[PW_OFF + (v - VOCAB)] = red[0];
  }
}

// ---------------- Kernel A2: per-position scores, softmax over K=4
__global__ void __launch_bounds__(256) k_scores(const int* __restrict__ x,
                                                const unsigned char* __restrict__ mask,
                                                const float* __restrict__ score_b,
                                                float* __restrict__ ws) {
  int gid = blockIdx.x * blockDim.x + threadIdx.x;
  if (gid >= BB * NN) return;
  int b = gid / NN, l = gid % NN;
  const float* ew = ws + EW_OFF;
  const float* pw = ws + PW_OFF;
  float sb = score_b[0];
  float sc[4];
#pragma unroll
  for (int k = 0; k < 4; ++k) {
    int bs = k + 1;
    int lb = (l / bs) * bs;
    float sum = 0.f;
    int cnt = 0;
    for (int j = 0; j < bs; ++j) {
      int idx = b * NN + lb + j;
      if (mask[idx]) { sum += ew[x[idx]] + pw[j]; cnt++; }
    }
    float mean = (cnt > 0) ? (sum / (float)cnt) : 0.f;
    sc[k] = mean + sb;
  }
  float m = fmaxf(fmaxf(sc[0], sc[1]), fmaxf(sc[2], sc[3]));
  float e0 = __expf(sc[0] - m), e1 = __expf(sc[1] - m);
  float e2 = __expf(sc[2] - m), e3 = __expf(sc[3] - m);
  float inv = 1.0f / (e0 + e1 + e2 + e3);
  float4* s = (float4*)(ws + S_OFF);
  s[gid] = make_float4(e0 * inv, e1 * inv, e2 * inv, e3 * inv);
}

// ---------------- Kernel B: s2 = softmax(S S^T) S, streamed via WMMA f32 16x16x4
// One wave (32 lanes) per (batch, query tile). Inner K-dim = 4 matches the ISA shape.
__global__ void __launch_bounds__(32) k_attn(float* __restrict__ ws) {
  const float* s = ws + S_OFF;
  float* s2 = ws + S2_OFF;
  int qt = blockIdx.x % NT;
  int b  = blockIdx.x / NT;
  const float* sb = s + (size_t)b * NN * 4;
  int lane = threadIdx.x;
  int half = lane >> 4;   // 0 or 1
  int li   = lane & 15;

  // B operand: query tile, lane li supplies Q[li, 2*half + {0,1}]
  v2f qv;
  {
    const float2 q = *(const float2*)(sb + (qt * 16 + li) * 4 + half * 2);
    qv.x = q.x; qv.y = q.y;
  }
  float num0 = 0.f, num1 = 0.f, num2 = 0.f, num3 = 0.f, den = 0.f;

  for (int kt = 0; kt < NT; ++kt) {
    // A operand: key tile, same packing
    v2f kv;
    const float2 kd = *(const float2*)(sb + (kt * 16 + li) * 4 + half * 2);
    kv.x = kd.x; kv.y = kd.y;
    v8f c = {0.f, 0.f, 0.f, 0.f, 0.f, 0.f, 0.f, 0.f};
    // D[M=key_row, N=query_col]; lane (li,half) gets rows half*8+r, col li
    v8f g = __builtin_amdgcn_wmma_f32_16x16x4_f32(
        /*neg_a=*/false, kv, /*neg_b=*/false, qv,
        /*c_mod=*/(short)0, c, /*reuse_a=*/false, /*reuse_b=*/false);
    int tbase = kt * 16 + half * 8;
#pragma unroll
    for (int r = 0; r < 8; ++r) {
      float e = __expf(g[r]);                 // dots in [0,1] -> safe
      den += e;
      const float4 t = *(const float4*)(sb + (tbase + r) * 4);
      num0 += e * t.x; num1 += e * t.y; num2 += e * t.z; num3 += e * t.w;
    }
  }
  // fold the two key-halves (lane n holds keys half*8..half*8+7 for query li)
  num0 += __shfl_xor(num0, 16);
  num1 += __shfl_xor(num1, 16);
  num2 += __shfl_xor(num2, 16);
  num3 += __shfl_xor(num3, 16);
  den  += __shfl_xor(den, 16);
  if (lane < 16) {
    float inv = 1.0f / den;
    float4* o = (float4*)(s2 + ((size_t)b * NN + qt * 16 + li) * 4);
    *o = make_float4(num0 * inv, num1 * inv, num2 * inv, num3 * inv);
  }
}

// ---------------- Kernel C: recombine block reprs with s2, downsample by 4
// One block per (batch, 12-position chunk): 12 = lcm(1,2,3,4), DS-aligned -> 3 output rows.
__global__ void __launch_bounds__(256) k_out(const int* __restrict__ x,
                                             const unsigned char* __restrict__ mask,
                                             const float* __restrict__ token_emb,
                                             const float* __restrict__ pos_emb,
                                             const float* __restrict__ ws,
                                             float* __restrict__ out) {
  __shared__ int   sx[12];
  __shared__ float smk[12];
  __shared__ float ss2[48];
  int b = blockIdx.x / NCHUNK;
  int t = blockIdx.x % NCHUNK;
  int l0 = t * 12;
  if (threadIdx.x < 12) {
    sx[threadIdx.x]  = x[b * NN + l0 + threadIdx.x];
    smk[threadIdx.x] = mask[b * NN + l0 + threadIdx.x] ? 1.f : 0.f;
  }
  if (threadIdx.x < 48) {
    ss2[threadIdx.x] = (ws + S2_OFF)[((size_t)b * NN + l0) * 4 + threadIdx.x];
  }
  __syncthreads();

  float* out_mask = out + (size_t)BB * GROUPS * DD;
  for (int dd = 0; dd < 2; ++dd) {
    int d = threadIdx.x + dd * 256;
    float pe[4];
#pragma unroll
    for (int j = 0; j < 4; ++j) pe[j] = pos_emb[j * DD + d];
    float h[12];
#pragma unroll
    for (int r = 0; r < 12; ++r) h[r] = token_emb[(size_t)sx[r] * DD + d];

    float br1[12], br2[6], br3[4], br4[3];
#pragma unroll
    for (int r = 0; r < 12; ++r) br1[r] = smk[r] * (h[r] + pe[0]);
#pragma unroll
    for (int r = 0; r < 6; ++r) {
      float c = smk[2 * r] + smk[2 * r + 1];
      float sum = smk[2 * r] * (h[2 * r] + pe[0]) + smk[2 * r + 1] * (h[2 * r + 1] + pe[1]);
      br2[r] = (c > 0.f) ? sum / c : 0.f;
    }
#pragma unroll
    for (int r = 0; r < 4; ++r) {
      float c = 0.f, sum = 0.f;
#pragma unroll
      for (int j = 0; j < 3; ++j) { c += smk[3 * r + j]; sum += smk[3 * r + j] * (h[3 * r + j] + pe[j]); }
      br3[r] = (c > 0.f) ? sum / c : 0.f;
    }
#pragma unroll
    for (int r = 0; r < 3; ++r) {
      float c = 0.f, sum = 0.f;
#pragma unroll
      for (int j = 0; j < 4; ++j) { c += smk[4 * r + j]; sum += smk[4 * r + j] * (h[4 * r + j] + pe[j]); }
      br4[r] = (c > 0.f) ? sum / c : 0.f;
    }

#pragma unroll
    for (int g = 0; g < 3; ++g) {
      float acc = 0.f, cnt = 0.f;
#pragma unroll
      for (int q = 0; q < 4; ++q) {
        int l = g * 4 + q;
        float op = ss2[l * 4 + 0] * br1[l] + ss2[l * 4 + 1] * br2[l >> 1] +
                   ss2[l * 4 + 2] * br3[l / 3] + ss2[l * 4 + 3] * br4[l >> 2];
        acc += smk[l] * op;
        cnt += smk[l];
      }
      float o = (cnt > 0.f) ? acc / cnt : 0.f;
      out[((size_t)b * GROUPS + t * 3 + g) * DD + d] = o;
    }
  }
  if (threadIdx.x < 3) {
    int g = threadIdx.x;
    float any = smk[4 * g] + smk[4 * g + 1] + smk[4 * g + 2] + smk[4 * g + 3];
    out_mask[(size_t)b * GROUPS + t * 3 + g] = (any > 0.f) ? 1.f : 0.f;
  }
}

extern "C" void kernel_launch(void* const* d_in, const int* in_sizes, int n_in,
                              void* d_out, int out_size, void* d_ws, size_t ws_size,
                              hipStream_t stream) {
  const int*           x         = (const int*)d_in[0];
  const unsigned char* mask      = (const unsigned char*)d_in[1];
  const float*         token_emb = (const float*)d_in[2];
  const float*         pos_emb   = (const float*)d_in[3];
  const float*         score_w   = (const float*)d_in[4];
  const float*         score_b   = (const float*)d_in[5];
  float* ws  = (float*)d_ws;
  float* out = (float*)d_out;

  k_rowdots<<<VOCAB + MBS, 128, 0, stream>>>(token_emb, pos_emb, score_w, ws);
  k_scores<<<(BB * NN + 255) / 256, 256, 0, stream>>>(x, mask, score_b, ws);
  k_attn<<<BB * NT, 32, 0, stream>>>(ws);
  k_out<<<BB * NCHUNK, 256, 0, stream>>>(x, mask, token_emb, pos_emb, ws, out);
}